// FSQ_28381143892079
// MI455X (gfx1250) — compile-verified
//
#include <hip/hip_runtime.h>

typedef float v2f __attribute__((ext_vector_type(2)));
typedef float v4f __attribute__((ext_vector_type(4)));
typedef float v8f __attribute__((ext_vector_type(8)));

#define DMODEL 1024
#define NLV 8
#define ROWS_TOTAL (8 * 4096)
#define HALF_L 3.996f      /* L*(1-EPS)/2 = 8*0.999/2 */
#define LN_EPS 1e-5f

__device__ __forceinline__ float fast_tanh(float z) {
#if __has_builtin(__builtin_amdgcn_tanhf)
    return __builtin_amdgcn_tanhf(z);   // native V_TANH_F32 on gfx1250
#else
    return tanhf(z);
#endif
}

__launch_bounds__(256, 2)
__global__ void fsq_wmma_kernel(const float* __restrict__ x,
                                const float* __restrict__ lnw,
                                const float* __restrict__ lnb,
                                const float* __restrict__ W,
                                float* __restrict__ out) {
    const int lane = threadIdx.x & 31;
    const int wave = threadIdx.x >> 5;
    const int tile = blockIdx.x * 8 + wave;   // one 16-row tile per wave32
    const int rowBase = tile * 16;

    // ---------------- Phase 1: per-row mean / rstd (coalesced stream) -------
    float myMean = 0.0f, myRstd = 1.0f;
    for (int r = 0; r < 16; ++r) {
        const float* rp = x + (size_t)(rowBase + r) * DMODEL;
        float s = 0.0f, ss = 0.0f;
        #pragma unroll
        for (int j = 0; j < 8; ++j) {
            v4f v = *(const v4f*)(rp + j * 128 + lane * 4);
            s  += v.x + v.y + v.z + v.w;
            ss += v.x * v.x + v.y * v.y + v.z * v.z + v.w * v.w;
        }
        #pragma unroll
        for (int m = 16; m >= 1; m >>= 1) {
            s  += __shfl_xor(s,  m, 32);
            ss += __shfl_xor(ss, m, 32);
        }
        const float mean = s * (1.0f / DMODEL);
        const float var  = ss * (1.0f / DMODEL) - mean * mean;
        const float rstd = rsqrtf(var + LN_EPS);
        // A-fragment layout: lanes L and L+16 both own matrix row (L&15)
        if ((lane & 15) == r) { myMean = mean; myRstd = rstd; }
    }

    // ---------------- Phase 2: WMMA k-loop (L2-resident re-read) ------------
    // A (16x4 f32): lanes 0-15 -> K{0,1} in VGPR{0,1}; lanes 16-31 -> K{2,3}.
    // B (4x16 f32): lanes 0-15 -> K rows {0,1}, N=lane; lanes 16-31 -> K{2,3}.
    const int hi = lane >> 4;          // 0: K-pair {0,1}, 1: K-pair {2,3}
    const int n  = lane & 15;          // output column (valid for n < 8)
    const bool nvalid = (n < NLV);

    const float* xrow = x   + (size_t)(rowBase + (lane & 15)) * DMODEL + 2 * hi;
    const float* wrow = W   + (size_t)(n & 7) * DMODEL + 2 * hi;  // clamped, in-bounds
    const float* lwp  = lnw + 2 * hi;
    const float* lbp  = lnb + 2 * hi;

    v8f acc = {};
    #pragma unroll 4
    for (int k = 0; k < DMODEL / 4; ++k) {
        const int c = k * 4;
        v2f xv = *(const v2f*)(xrow + c);
        v2f wv = *(const v2f*)(wrow + c);
        v2f lw = *(const v2f*)(lwp + c);
        v2f lb = *(const v2f*)(lbp + c);

        v2f a, b;
        a.x = (xv.x - myMean) * myRstd * lw.x + lb.x;
        a.y = (xv.y - myMean) * myRstd * lw.y + lb.y;
        b.x = nvalid ? wv.x : 0.0f;
        b.y = nvalid ? wv.y : 0.0f;

        // D(16x16,f32) += A(16x4,f32) x B(4x16,f32)
        acc = __builtin_amdgcn_wmma_f32_16x16x4_f32(
            /*neg_a=*/false, a, /*neg_b=*/false, b,
            /*c_mod=*/(short)0, acc, /*reuse_a=*/false, /*reuse_b=*/false);
    }

    // ---------------- Epilogue: tanh bound + RNE round, store [rows x 8] ----
    // C/D layout: VGPR v, lane L -> row (v + 8*(L/16)), col (L&15).
    #pragma unroll
    for (int v = 0; v < 8; ++v) {
        const int row = rowBase + v + 8 * hi;
        const float bounded = HALF_L * fast_tanh(acc[v]);
        const float q = rintf(bounded);   // round-half-to-even, matches jnp.round
        if (nvalid) {
            out[(size_t)row * NLV + n] = q;
        }
    }
}

extern "C" void kernel_launch(void* const* d_in, const int* in_sizes, int n_in,
                              void* d_out, int out_size, void* d_ws, size_t ws_size,
                              hipStream_t stream) {
    const float* regrs = (const float*)d_in[0];   // [8,4096,1024] f32
    const float* ln_w  = (const float*)d_in[1];   // [1024] f32
    const float* ln_b  = (const float*)d_in[2];   // [1024] f32
    const float* W     = (const float*)d_in[3];   // [8,1024] f32
    float* out = (float*)d_out;                   // [8,4096,8] f32

    (void)in_sizes; (void)n_in; (void)out_size; (void)d_ws; (void)ws_size;

    dim3 block(256);                       // 8 wave32s -> 8 tiles of 16 rows
    dim3 grid(ROWS_TOTAL / (16 * 8));      // 256 blocks
    hipLaunchKernelGGL(fsq_wmma_kernel, grid, block, 0, stream,
                       regrs, ln_w, ln_b, W, out);
}